// MultiVariationalGCN_21904333209748
// MI455X (gfx1250) — compile-verified
//
#include <hip/hip_runtime.h>

#define NN 100000
#define NE 1600000

typedef float v2f __attribute__((ext_vector_type(2)));
typedef float v8f __attribute__((ext_vector_type(8)));

// ---------------- degree / normalization ----------------
__global__ void deg_init_k(float* deg, int n) {
  int i = blockIdx.x * blockDim.x + threadIdx.x;
  if (i < n) deg[i] = 1.0f;                      // self-loop weight
}

__global__ void deg_accum_k(const int* __restrict__ row, const float* __restrict__ w,
                            float* deg, int e) {
  int i = blockIdx.x * blockDim.x + threadIdx.x;
  if (i < e) atomicAdd(&deg[row[i]], w[i]);
}

__global__ void deg_rsqrt_k(float* deg, int n) {
  int i = blockIdx.x * blockDim.x + threadIdx.x;
  if (i < n) deg[i] = rsqrtf(deg[i]);            // deg >= 1 always (self-loop)
}

__global__ void edge_norm_k(const int* __restrict__ row, const int* __restrict__ col,
                            const float* __restrict__ w, const float* __restrict__ dis,
                            float* __restrict__ nrm, int e) {
  int i = blockIdx.x * blockDim.x + threadIdx.x;
  if (i < e) nrm[i] = dis[row[i]] * w[i] * dis[col[i]];
}

// ---------------- weight transpose: Wt[U][K] = W[K][U], K = 128 ----------------
__global__ void transpose_k(const float* __restrict__ W, float* __restrict__ Wt, int U) {
  int i = blockIdx.x * blockDim.x + threadIdx.x;   // over 128*U
  if (i >= 128 * U) return;
  int k = i / U, u = i % U;
  Wt[(size_t)u * 128 + k] = W[i];
}

// ---------------- WMMA fp32 GEMM: T[M,U] = act(X[M,128]) @ Wt^T ----------------
// One wave per 16x16 output tile; V_WMMA_F32_16X16X4_F32, K unrolled 32x.
// A (16x4 f32): lane L holds row M=L&15, Ks {2*(L>>4), 2*(L>>4)+1} in {v0,v1}.
// B (4x16 f32): lane L holds col N=L&15, same K pair mapping (contiguous in Wt).
// C/D (16x16 f32): vgpr r, lane L -> M = r + 8*(L>>4), N = L&15.
template <int U, bool RELU_IN>
__global__ __launch_bounds__(128) void gemm_wmma_k(const float* __restrict__ X,
                                                   const float* __restrict__ Wt,
                                                   float* __restrict__ T, int M) {
  const int lane   = threadIdx.x & 31;
  const int wave   = threadIdx.x >> 5;
  const int tile_m = blockIdx.x * 4 + wave;
  const int tile_n = blockIdx.y;
  if (tile_m * 16 >= M) return;                  // wave-uniform: EXEC all-1s inside

  const int mrow = tile_m * 16 + (lane & 15);
  const int ncol = tile_n * 16 + (lane & 15);
  const int kk   = (lane >> 4) * 2;

  const float* xr = X + (size_t)mrow * 128 + kk;   // 8B-aligned
  const float* wr = Wt + (size_t)ncol * 128 + kk;  // 8B-aligned

  v8f c = {0.f, 0.f, 0.f, 0.f, 0.f, 0.f, 0.f, 0.f};
#pragma unroll
  for (int k = 0; k < 128; k += 4) {
    v2f a = *reinterpret_cast<const v2f*>(xr + k);
    if (RELU_IN) { a.x = fmaxf(a.x, 0.0f); a.y = fmaxf(a.y, 0.0f); }
    v2f b = *reinterpret_cast<const v2f*>(wr + k);
    c = __builtin_amdgcn_wmma_f32_16x16x4_f32(false, a, false, b, (short)0, c,
                                              false, false);
  }

  const int mbase = tile_m * 16 + (lane >> 4) * 8;
  float* trow = T + (size_t)mbase * U + ncol;
#pragma unroll
  for (int r = 0; r < 8; ++r) trow[(size_t)r * U] = c[r];
}

// ---------------- aggregation: out = b + dis[i]^2 * T[i] + scatter ----------------
template <int UV>
__global__ void agg_init_k(const float4* __restrict__ T, const float* __restrict__ dis,
                           const float* __restrict__ bias, float4* __restrict__ out,
                           int n) {
  long long idx = (long long)blockIdx.x * blockDim.x + threadIdx.x;
  if (idx >= (long long)n * UV) return;
  int i  = (int)(idx / UV);
  int u4 = (int)(idx % UV);
  float d  = dis[i];
  float sn = d * d;                              // self-loop norm = dis^2
  float4 t = T[idx];
  float4 b = reinterpret_cast<const float4*>(bias)[u4];
  float4 o;
  o.x = b.x + sn * t.x; o.y = b.y + sn * t.y;
  o.z = b.z + sn * t.z; o.w = b.w + sn * t.w;
  out[idx] = o;
}

template <int UV>
__global__ void agg_scatter_k(const int* __restrict__ row, const int* __restrict__ col,
                              const float* __restrict__ nrm, const float* __restrict__ T,
                              float* out, int e) {
  long long idx = (long long)blockIdx.x * blockDim.x + threadIdx.x;
  if (idx >= (long long)e * UV) return;
  int ei = (int)(idx / UV);
  int u4 = (int)(idx % UV) * 4;
  int r = row[ei], cc = col[ei];
  float w = nrm[ei];
  const int Uu = UV * 4;
  const float4 t = *reinterpret_cast<const float4*>(T + (size_t)cc * Uu + u4);
  float* dst = out + (size_t)r * Uu + u4;
  atomicAdd(dst + 0, w * t.x);
  atomicAdd(dst + 1, w * t.y);
  atomicAdd(dst + 2, w * t.z);
  atomicAdd(dst + 3, w * t.w);
}

// ---------------- reparameterize + KL ----------------
__global__ void final_k(const float* __restrict__ rnd, float* out, int nz) {
  int i = blockIdx.x * blockDim.x + threadIdx.x;
  if (i >= nz) return;
  float mu = out[i];
  float ls = out[nz + i];
  float s  = expf(ls);
  out[i]      = mu + rnd[i] * s;                          // h_out
  out[nz + i] = -ls + 0.5f * (s * s + mu * mu - 1.0f);    // kl
}

extern "C" void kernel_launch(void* const* d_in, const int* in_sizes, int n_in,
                              void* d_out, int out_size, void* d_ws, size_t ws_size,
                              hipStream_t stream) {
  (void)in_sizes; (void)n_in; (void)out_size; (void)ws_size;
  const int N = NN, E = NE, H = 128, DZ = 64;
  const float* h   = (const float*)d_in[0];
  const int*   ei  = (const int*)d_in[1];
  const float* ew  = (const float*)d_in[2];
  const float* rnd = (const float*)d_in[3];
  const float* W0  = (const float*)d_in[4];
  const float* b0  = (const float*)d_in[5];
  const float* W1  = (const float*)d_in[6];
  const float* b1  = (const float*)d_in[7];
  const float* Wmu = (const float*)d_in[8];
  const float* bmu = (const float*)d_in[9];
  const float* Wsd = (const float*)d_in[10];
  const float* bsd = (const float*)d_in[11];
  const int* row = ei;           // edge_index[0]
  const int* col = ei + E;       // edge_index[1]
  float* out = (float*)d_out;
  const int NZ = N * DZ;

  // workspace carve-up (~135 MB of float32 + ~200 KB transposed weights)
  float* ws    = (float*)d_ws;
  float* dis   = ws;                            // N
  float* nrm   = dis + N;                       // E
  float* bufT  = nrm + E;                       // N*128 (transform result)
  float* bufH  = bufT + (size_t)N * 128;        // N*128 (aggregated hidden)
  float* bufS  = bufH + (size_t)N * 128;        // N*64  (mu/std transform)
  float* Wt0   = bufS + (size_t)N * 64;         // 128*128
  float* Wt1   = Wt0 + 128 * 128;               // 128*128
  float* Wtmu  = Wt1 + 128 * 128;               // 64*128
  float* Wtsd  = Wtmu + 64 * 128;               // 64*128

  const int TB = 256;
  deg_init_k <<<(N + TB - 1) / TB, TB, 0, stream>>>(dis, N);
  deg_accum_k<<<(E + TB - 1) / TB, TB, 0, stream>>>(row, ew, dis, E);
  deg_rsqrt_k<<<(N + TB - 1) / TB, TB, 0, stream>>>(dis, N);
  edge_norm_k<<<(E + TB - 1) / TB, TB, 0, stream>>>(row, col, ew, dis, nrm, E);

  transpose_k<<<(128 * H  + TB - 1) / TB, TB, 0, stream>>>(W0,  Wt0,  H);
  transpose_k<<<(128 * H  + TB - 1) / TB, TB, 0, stream>>>(W1,  Wt1,  H);
  transpose_k<<<(128 * DZ + TB - 1) / TB, TB, 0, stream>>>(Wmu, Wtmu, DZ);
  transpose_k<<<(128 * DZ + TB - 1) / TB, TB, 0, stream>>>(Wsd, Wtsd, DZ);

  const int MT = N / 16;                       // 6250 tiles in M (N % 16 == 0)
  dim3 gblk(128);                              // 4 waves / block
  const unsigned MB = (unsigned)((MT + 3) / 4);

  // ---- layer 1: bufT = h @ W0 ; bufH = aggregate(bufT) + b0 (ReLU folded fwd)
  gemm_wmma_k<128, false><<<dim3(MB, H / 16), gblk, 0, stream>>>(h, Wt0, bufT, N);
  {
    long long ti = (long long)N * 32, ts = (long long)E * 32;
    agg_init_k<32>   <<<(unsigned)((ti + TB - 1) / TB), TB, 0, stream>>>((const float4*)bufT, dis, b0, (float4*)bufH, N);
    agg_scatter_k<32><<<(unsigned)((ts + TB - 1) / TB), TB, 0, stream>>>(row, col, nrm, bufT, bufH, E);
  }
  // ---- layer 2: bufT = relu(bufH) @ W1 ; bufH = aggregate(bufT) + b1
  gemm_wmma_k<128, true><<<dim3(MB, H / 16), gblk, 0, stream>>>(bufH, Wt1, bufT, N);
  {
    long long ti = (long long)N * 32, ts = (long long)E * 32;
    agg_init_k<32>   <<<(unsigned)((ti + TB - 1) / TB), TB, 0, stream>>>((const float4*)bufT, dis, b1, (float4*)bufH, N);
    agg_scatter_k<32><<<(unsigned)((ts + TB - 1) / TB), TB, 0, stream>>>(row, col, nrm, bufT, bufH, E);
  }
  // ---- mu head: bufS = relu(bufH) @ Wmu ; out[0:NZ] = aggregate(bufS) + bmu
  gemm_wmma_k<64, true><<<dim3(MB, DZ / 16), gblk, 0, stream>>>(bufH, Wtmu, bufS, N);
  {
    long long ti = (long long)N * 16, ts = (long long)E * 16;
    agg_init_k<16>   <<<(unsigned)((ti + TB - 1) / TB), TB, 0, stream>>>((const float4*)bufS, dis, bmu, (float4*)out, N);
    agg_scatter_k<16><<<(unsigned)((ts + TB - 1) / TB), TB, 0, stream>>>(row, col, nrm, bufS, out, E);
  }
  // ---- log_std head: bufS = relu(bufH) @ Wstd ; out[NZ:2NZ] = aggregate(bufS) + bstd
  gemm_wmma_k<64, true><<<dim3(MB, DZ / 16), gblk, 0, stream>>>(bufH, Wtsd, bufS, N);
  {
    long long ti = (long long)N * 16, ts = (long long)E * 16;
    agg_init_k<16>   <<<(unsigned)((ti + TB - 1) / TB), TB, 0, stream>>>((const float4*)bufS, dis, bsd, (float4*)(out + NZ), N);
    agg_scatter_k<16><<<(unsigned)((ts + TB - 1) / TB), TB, 0, stream>>>(row, col, nrm, bufS, out + NZ, E);
  }
  // ---- reparameterized sample + KL, in place over d_out
  final_k<<<(NZ + TB - 1) / TB, TB, 0, stream>>>(rnd, out, NZ);
}